// ProposalGenerator_34746285425347
// MI455X (gfx1250) — compile-verified
//
#include <hip/hip_runtime.h>
#include <hip/hip_bf16.h>

typedef __attribute__((ext_vector_type(2))) float v2f;
typedef __attribute__((ext_vector_type(8))) float v8f;

#define B_  16
#define N_  16384
#define C_  256
#define NP_ 128
#define NS_ 16
#define R_  0.3f

// d_out layout (floats): [0,6144) new_xyz (B,NP,3) | [6144,268288) feat (B,128,NP) |
// [268288,270336) fps_inds (B,NP) as int32 bits.
#define OUT_FEAT_OFF 6144
#define OUT_IND_OFF  (6144 + 262144)

// ---------------------------------------------------------------------------
// Kernel 1: farthest point sampling. One workgroup per batch.
// Entire point cloud (SoA) + min-dist array live in LDS (256 KB < 320 KB WGP LDS).
// ---------------------------------------------------------------------------
__global__ __launch_bounds__(1024) void fps_kernel(const float* __restrict__ vxyz,
                                                   float* __restrict__ out) {
  extern __shared__ float sm[];
  float* xs   = sm;            // N
  float* ys   = xs + N_;       // N
  float* zs   = ys + N_;       // N
  float* mind = zs + N_;       // N
  float* rv   = mind + N_;     // 32
  int*   ri   = (int*)(rv + 32);
  int*   slast = ri + 32;

  const int b    = blockIdx.x;
  const int tid  = threadIdx.x;
  const int nthr = blockDim.x;
  const int lane = tid & 31;
  const int wv   = tid >> 5;
  const int nw   = nthr >> 5;

  const float* px = vxyz + (long)b * N_ * 3;
  for (int i = tid; i < N_; i += nthr) {
    xs[i] = px[i * 3 + 0];
    ys[i] = px[i * 3 + 1];
    zs[i] = px[i * 3 + 2];
    mind[i] = 1e10f;
  }
  if (tid == 0) slast[0] = 0;
  __syncthreads();

  int* out_ind = (int*)(out + OUT_IND_OFF);
  if (tid == 0) {
    out_ind[b * NP_] = 0;
    out[(b * NP_) * 3 + 0] = xs[0];
    out[(b * NP_) * 3 + 1] = ys[0];
    out[(b * NP_) * 3 + 2] = zs[0];
  }

  for (int it = 1; it < NP_; ++it) {
    const int last = slast[0];
    const float lx = xs[last], ly = ys[last], lz = zs[last];
    float bv = -1.0f;
    int   bi = 0;
    for (int i = tid; i < N_; i += nthr) {
      const float dx = xs[i] - lx, dy = ys[i] - ly, dz = zs[i] - lz;
      const float d = dx * dx + dy * dy + dz * dz;
      const float m = fminf(mind[i], d);
      mind[i] = m;
      if (m > bv) { bv = m; bi = i; }
    }
    // wave32 argmax (smallest index on ties, matching argmax-first semantics)
    for (int off = 16; off > 0; off >>= 1) {
      const float ov = __shfl_down(bv, off, 32);
      const int   oi = __shfl_down(bi, off, 32);
      if (ov > bv || (ov == bv && oi < bi)) { bv = ov; bi = oi; }
    }
    if (lane == 0) { rv[wv] = bv; ri[wv] = bi; }
    __syncthreads();
    if (wv == 0) {
      float v = (lane < nw) ? rv[lane] : -1.0f;
      int   i2 = (lane < nw) ? ri[lane] : 0;
      for (int off = 16; off > 0; off >>= 1) {
        const float ov = __shfl_down(v, off, 32);
        const int   oi = __shfl_down(i2, off, 32);
        if (ov > v || (ov == v && oi < i2)) { v = ov; i2 = oi; }
      }
      if (lane == 0) {
        slast[0] = i2;
        out_ind[b * NP_ + it] = i2;
        out[(b * NP_ + it) * 3 + 0] = xs[i2];
        out[(b * NP_ + it) * 3 + 1] = ys[i2];
        out[(b * NP_ + it) * 3 + 2] = zs[i2];
      }
    }
    __syncthreads();
  }
}

// ---------------------------------------------------------------------------
// Kernel 2: ball query. One wave32 per center; ballot + prefix-popcount append
// keeps the first NS in-ball indices in ascending order (matches the sort).
// ---------------------------------------------------------------------------
__global__ __launch_bounds__(256) void ballq_kernel(const float* __restrict__ vxyz,
                                                    const float* __restrict__ out,
                                                    int* __restrict__ wsidx) {
  __shared__ int sel[8][NS_];
  const int tid  = threadIdx.x;
  const int lane = tid & 31;
  const int wv   = tid >> 5;
  const int g    = blockIdx.x * 8 + wv;  // global center
  const int b    = g / NP_;

  const float cx = out[g * 3 + 0], cy = out[g * 3 + 1], cz = out[g * 3 + 2];
  const float* px = vxyz + (long)b * N_ * 3;
  const float r2 = R_ * R_;

  int cnt = 0;
  for (int j0 = 0; j0 < N_; j0 += 32) {
    const int j = j0 + lane;
    const float dx = px[j * 3 + 0] - cx;
    const float dy = px[j * 3 + 1] - cy;
    const float dz = px[j * 3 + 2] - cz;
    const bool pred = (dx * dx + dy * dy + dz * dz) < r2;
    const unsigned m = (unsigned)__ballot(pred);
    const int pre = __popc(m & ((1u << lane) - 1u));
    if (pred && (cnt + pre) < NS_) sel[wv][cnt + pre] = j;
    cnt += __popc(m);
    if (cnt >= NS_) break;
  }
  if (cnt == 0) {
    if (lane < NS_) sel[wv][lane] = 0;
  } else if (cnt < NS_) {
    if (lane >= cnt && lane < NS_) sel[wv][lane] = sel[wv][0];
  }
  if (lane < NS_) wsidx[g * NS_ + lane] = sel[wv][lane];
}

// ---------------------------------------------------------------------------
// Kernel 3: fused grouping + 3 conv layers (WMMA f32 16x16x4) + max-pool +
// 2 FC layers (cooperative WMMA GEMM over the block's 8 centers).
// ---------------------------------------------------------------------------
__global__ __launch_bounds__(256) void group_mlp_kernel(
    const float* __restrict__ vxyz, const float* __restrict__ feats,
    const float* __restrict__ W1, const float* __restrict__ W2,
    const float* __restrict__ W3, const float* __restrict__ bnS,
    const float* __restrict__ bnB, const float* __restrict__ Wm,
    const float* __restrict__ bm, const float* __restrict__ mS,
    const float* __restrict__ mB, const int* __restrict__ wsidx,
    float* out) {
  extern __shared__ float sm[];
  v2f*   wpair = (v2f*)sm;                 // up to 130*128 v2f (133 KB), reused per layer
  float* hbuf  = sm + 130 * 128 * 2;       // 8 waves * 128 * 16 floats (64 KB)
  float* fmat  = hbuf + 8 * 128 * 16;      // 128 * 16 floats (8 KB)
  int*   sidx  = (int*)(fmat + 128 * 16);  // 8 * 16
  float* cxyz  = (float*)(sidx + 8 * 16);  // 8 * 3

  const int tid  = threadIdx.x;
  const int lane = tid & 31;
  const int wv   = tid >> 5;
  const int half = lane >> 4;   // 0: lanes 0-15, 1: lanes 16-31
  const int s    = lane & 15;   // sample / column index
  const int g0   = blockIdx.x * 8;
  const int b    = g0 / NP_;

  if (tid < 128) sidx[tid] = wsidx[g0 * NS_ + tid];
  if (tid < 24)  cxyz[tid] = out[g0 * 3 + tid];
  // stage W1 as K-pair fragments: wpair[p*128+o] = {W1[o][2p], W1[o][2p+1]}, K padded to 260
  for (int p = tid; p < 130 * 128; p += blockDim.x) {
    const int pr = p >> 7, o = p & 127;
    const int c0 = pr * 2, c1 = pr * 2 + 1;
    v2f w;
    w.x = (c0 < 259) ? W1[o * 259 + c0] : 0.0f;
    w.y = (c1 < 259) ? W1[o * 259 + c1] : 0.0f;
    wpair[p] = w;
  }
  __syncthreads();

  const int  pidx = sidx[wv * NS_ + s];
  const long fb   = (long)b * C_ * N_;
  const float invR = 1.0f / R_;
  float gx, gy, gz;
  {
    const float* pp = vxyz + ((long)b * N_ + pidx) * 3;
    gx = (pp[0] - cxyz[wv * 3 + 0]) * invR;
    gy = (pp[1] - cxyz[wv * 3 + 1]) * invR;
    gz = (pp[2] - cxyz[wv * 3 + 2]) * invR;
  }

  v8f acc[8];
  #pragma unroll
  for (int m = 0; m < 8; ++m)
    #pragma unroll
    for (int r = 0; r < 8; ++r) acc[m][r] = 0.0f;

  // ---- layer 1: D(128x16) += W1(128x260) x g(260x16), K-steps of 4 ----
  for (int k = 0; k < 65; ++k) {
    v2f bf;
    if (k == 0) {                         // g channels 0..3 = {gx,gy,gz,feat0}
      if (half == 0) { bf.x = gx; bf.y = gy; }
      else           { bf.x = gz; bf.y = feats[fb + pidx]; }
    } else if (k < 64) {                  // channels 4k..4k+3 -> feature 4k-3..4k
      const int ch = 4 * k + 2 * half - 3;
      bf.x = feats[fb + (long)ch * N_ + pidx];
      bf.y = feats[fb + (long)(ch + 1) * N_ + pidx];
    } else {                              // channels 256..259 (259 = zero pad)
      const int ch = 253 + 2 * half;
      bf.x = feats[fb + (long)ch * N_ + pidx];
      bf.y = (half == 0) ? feats[fb + (long)(ch + 1) * N_ + pidx] : 0.0f;
    }
    const v2f* wrow = wpair + (2 * k + half) * 128 + s;
    #pragma unroll
    for (int m = 0; m < 8; ++m) {
      const v2f af = wrow[m * 16];
      acc[m] = __builtin_amdgcn_wmma_f32_16x16x4_f32(false, af, false, bf,
                                                     (short)0, acc[m], false, false);
    }
  }

  float* hw = hbuf + wv * 128 * 16;
  #pragma unroll
  for (int m = 0; m < 8; ++m)
    #pragma unroll
    for (int r = 0; r < 8; ++r) {
      const int o = m * 16 + r + half * 8;
      hw[o * 16 + s] = fmaxf(acc[m][r] * bnS[o] + bnB[o], 0.0f);
    }

  // ---- layers 2 & 3: 128->128 ----
  const float* Ws[2] = {W2, W3};
  for (int L = 0; L < 2; ++L) {
    __syncthreads();
    const float* W = Ws[L];
    for (int p = tid; p < 64 * 128; p += blockDim.x) {
      const int pr = p >> 7, o = p & 127;
      v2f w; w.x = W[o * 128 + pr * 2]; w.y = W[o * 128 + pr * 2 + 1];
      wpair[p] = w;
    }
    __syncthreads();
    #pragma unroll
    for (int m = 0; m < 8; ++m)
      #pragma unroll
      for (int r = 0; r < 8; ++r) acc[m][r] = 0.0f;
    for (int k = 0; k < 32; ++k) {
      v2f bf;
      const int c0 = 4 * k + 2 * half;
      bf.x = hw[c0 * 16 + s];
      bf.y = hw[(c0 + 1) * 16 + s];
      const v2f* wrow = wpair + (2 * k + half) * 128 + s;
      #pragma unroll
      for (int m = 0; m < 8; ++m) {
        const v2f af = wrow[m * 16];
        acc[m] = __builtin_amdgcn_wmma_f32_16x16x4_f32(false, af, false, bf,
                                                       (short)0, acc[m], false, false);
      }
    }
    const float* sc = bnS + (L + 1) * 128;
    const float* bi = bnB + (L + 1) * 128;
    #pragma unroll
    for (int m = 0; m < 8; ++m)
      #pragma unroll
      for (int r = 0; r < 8; ++r) {
        const int o = m * 16 + r + half * 8;
        hw[o * 16 + s] = fmaxf(acc[m][r] * sc[o] + bi[o], 0.0f);
      }
  }

  // ---- max over NS into shared feature matrix (cols = 8 local centers) ----
  __syncthreads();
  for (int i = tid; i < 128 * 16; i += blockDim.x) fmat[i] = 0.0f;
  __syncthreads();
  #pragma unroll
  for (int j = 0; j < 4; ++j) {
    const int o = lane * 4 + j;
    float mv = hw[o * 16];
    #pragma unroll
    for (int n = 1; n < 16; ++n) mv = fmaxf(mv, hw[o * 16 + n]);
    fmat[o * 16 + wv] = mv;
  }

  // ---- FC layers: cooperative 128x16x128 GEMM, wave wv owns M-tile wv ----
  for (int L = 0; L < 2; ++L) {
    __syncthreads();
    const float* W = Wm + L * 128 * 128;
    for (int p = tid; p < 64 * 128; p += blockDim.x) {
      const int pr = p >> 7, o = p & 127;
      v2f w; w.x = W[o * 128 + pr * 2]; w.y = W[o * 128 + pr * 2 + 1];
      wpair[p] = w;
    }
    __syncthreads();
    v8f a2;
    #pragma unroll
    for (int r = 0; r < 8; ++r) a2[r] = 0.0f;
    for (int k = 0; k < 32; ++k) {
      v2f bf;
      const int c0 = 4 * k + 2 * half;
      bf.x = fmat[c0 * 16 + s];
      bf.y = fmat[(c0 + 1) * 16 + s];
      const v2f af = wpair[(2 * k + half) * 128 + wv * 16 + s];
      a2 = __builtin_amdgcn_wmma_f32_16x16x4_f32(false, af, false, bf,
                                                 (short)0, a2, false, false);
    }
    __syncthreads();  // everyone done reading fmat before it is rewritten
    const float* bb = bm + L * 128;
    const float* sc = mS + L * 128;
    const float* bi = mB + L * 128;
    if (L == 0) {
      #pragma unroll
      for (int r = 0; r < 8; ++r) {
        const int o = wv * 16 + r + half * 8;
        fmat[o * 16 + s] = fmaxf((a2[r] + bb[o]) * sc[o] + bi[o], 0.0f);
      }
    } else {
      if (s < 8) {  // only 8 valid center columns per block
        const int gc = g0 + s;
        const int ob = gc / NP_, op = gc % NP_;
        #pragma unroll
        for (int r = 0; r < 8; ++r) {
          const int o = wv * 16 + r + half * 8;
          const float v = fmaxf((a2[r] + bb[o]) * sc[o] + bi[o], 0.0f);
          out[OUT_FEAT_OFF + (long)ob * 128 * NP_ + o * NP_ + op] = v;
        }
      }
    }
  }
}

extern "C" void kernel_launch(void* const* d_in, const int* in_sizes, int n_in,
                              void* d_out, int out_size, void* d_ws, size_t ws_size,
                              hipStream_t stream) {
  const float* vxyz  = (const float*)d_in[0];
  const float* vfeat = (const float*)d_in[1];
  const float* W1    = (const float*)d_in[2];
  const float* W2    = (const float*)d_in[3];
  const float* W3    = (const float*)d_in[4];
  const float* bnS   = (const float*)d_in[5];
  const float* bnB   = (const float*)d_in[6];
  const float* Wm    = (const float*)d_in[7];
  const float* bm    = (const float*)d_in[8];
  const float* mS    = (const float*)d_in[9];
  const float* mB    = (const float*)d_in[10];
  float* out  = (float*)d_out;
  int* wsidx  = (int*)d_ws;  // B*NP*NS ints = 128 KB

  const size_t smem1 = (size_t)(4 * N_) * sizeof(float) + 512;  // 256.5 KB < 320 KB
  fps_kernel<<<B_, 1024, smem1, stream>>>(vxyz, out);

  ballq_kernel<<<(B_ * NP_) / 8, 256, 0, stream>>>(vxyz, out, wsidx);

  const size_t smem3 = (size_t)(130 * 128 * 2 + 8 * 128 * 16 + 128 * 16) * sizeof(float) +
                       (size_t)(8 * 16) * sizeof(int) + (size_t)(8 * 3 + 8) * sizeof(float);
  group_mlp_kernel<<<(B_ * NP_) / 8, 256, smem3, stream>>>(
      vxyz, vfeat, W1, W2, W3, bnS, bnB, Wm, bm, mS, mB, wsidx, out);
}